// frc_2b_86397562126884
// MI455X (gfx1250) — compile-verified
//
#include <hip/hip_runtime.h>
#include <math.h>

typedef __attribute__((ext_vector_type(2))) float v2f;
typedef __attribute__((ext_vector_type(8))) float v8f;

#define NB    4096
#define PAIRS 512
#define WDT   10
#define EPS2  1e-12f
#define RS3   0.57735026918962576f   // 1/sqrt(3)
#define RS10  0.31622776601683794f   // 1/sqrt(10)
#define SLABW 17                     // row stride (floats) -> conflict-free A reads
#define SLABSZ (16 * SLABW)          // one 16x16 matrix slab

__device__ __forceinline__ v8f wmma4(v2f a, v2f b, v8f c) {
    // V_WMMA_F32_16X16X4_F32 : D = A(16x4) * B(4x16) + C
    return __builtin_amdgcn_wmma_f32_16x16x4_f32(
        /*neg_a=*/false, a, /*neg_b=*/false, b,
        /*c_mod=*/(short)0, c, /*reuse_a=*/false, /*reuse_b=*/false);
}

__device__ __forceinline__ float fast_tanh(float x) {
#if __has_builtin(__builtin_amdgcn_tanhf)
    return __builtin_amdgcn_tanhf(x);
#else
    float r;
    // V_TANH_F32 is a TRANS op: 1 independent op / V_NOP before result use (ISA 7.4)
    asm("v_tanh_f32 %0, %1\n\tv_nop" : "=v"(r) : "v"(x));
    return r;
#endif
}

// B-fragment (4x16 chunk kb of a KxWDT row-major weight), zero-padded, pre-scaled.
// lane l: n = l&15, k0 = kb*4 + ((l>>4)<<1)
__device__ __forceinline__ v2f bfrag(const float* __restrict__ W, int K,
                                     int kb, int n, int k2, float scale) {
    int k0 = kb * 4 + k2;
    v2f r;
    r.x = (k0     < K && n < WDT) ? W[k0 * WDT + n] * scale : 0.0f;
    r.y = (k0 + 1 < K && n < WDT) ? W[(k0 + 1) * WDT + n] * scale : 0.0f;
    return r;
}

__global__ __launch_bounds__(256) void frc2b_kernel(
    const float* __restrict__ invar,   // [B,P,5]
    const float* __restrict__ vecs,    // [B,P,9]
    const float* __restrict__ w1_0,    // [5,10]
    const float* __restrict__ b1_0,    // [10]
    const float* __restrict__ w1_1,    // [10,10]
    const float* __restrict__ b1_1,    // [10]
    const float* __restrict__ lin0_w,  // [3,10]
    const float* __restrict__ na0_b,   // [10]
    const float* __restrict__ lin1_w,  // [10,10]
    const float* __restrict__ na1_b,   // [10]
    const float* __restrict__ final_w, // [10,1]
    float* __restrict__ out)           // [B,3]
{
    __shared__ float slab[8 * 3 * SLABSZ];  // 3 matrix slabs per wave
    __shared__ float red[8][3];

    const int tid  = threadIdx.x;
    const int l    = tid & 31;        // lane in wave32
    const int wv   = tid >> 5;        // wave 0..7
    const int b    = blockIdx.x;

    const int n    = l & 15;          // C/D column, A row (m), B column
    const int half = (l >> 4) & 1;    // lane-half select
    const int k2   = half << 1;       // k offset within a 4-wide K chunk
    const float hmask = half ? 0.0f : 1.0f;  // branch-free K padding mask

    float* myslab = &slab[wv * 3 * SLABSZ];

    // ---- weight fragments: load once, keep in VGPRs ----
    const v2f bw10_0 = bfrag(w1_0,   5, 0, n, k2, 1.0f);
    const v2f bw10_1 = bfrag(w1_0,   5, 1, n, k2, 1.0f);
    v2f bw11[3], blin1[3];
    #pragma unroll
    for (int kb = 0; kb < 3; ++kb) {
        bw11[kb]  = bfrag(w1_1,   WDT, kb, n, k2, 1.0f);
        blin1[kb] = bfrag(lin1_w, WDT, kb, n, k2, RS10);
    }
    const v2f blin0 = bfrag(lin0_w, 3, 0, n, k2, RS3);

    const float bb10 = (n < WDT) ? b1_0[n] : 0.0f;
    const float bb11 = (n < WDT) ? b1_1[n] : 0.0f;
    const float nb0  = (n < WDT) ? na0_b[n] : 0.0f;
    const float nb1  = (n < WDT) ? na1_b[n] : 0.0f;
    const float coef = (n < WDT) ? final_w[n] * RS10 : 0.0f;

    float acc0 = 0.0f, acc1 = 0.0f, acc2 = 0.0f;

    for (int t = 0; t < 4; ++t) {
        const int row0 = (wv * 4 + t) * 16;
        const float* ivp = invar + ((size_t)b * PAIRS + row0) * 5;
        const float* vvp = vecs  + ((size_t)b * PAIRS + row0) * 9;
        if (t < 3) {  // global_prefetch_b8 for next tile, keep near
            __builtin_prefetch(ivp + 16 * 5, 0, 3);
            __builtin_prefetch(vvp + 16 * 9, 0, 3);
        }

        // ===== scalar branch, stage 1: [16x5] @ [5x10], K padded to 8 =====
        // Branch-free A fragments: all addresses in-bounds, pad by hmask.
        v8f yc = {};
        {
            v2f a0, a1;
            a0.x = ivp[n * 5 + k2];            // k = {0,1}|{2,3}
            a0.y = ivp[n * 5 + k2 + 1];
            a1.x = ivp[n * 5 + 4] * hmask;     // k = 4 (half0), k = 6 -> 0 (half1)
            a1.y = 0.0f;                        // k = 5|7 -> 0
            yc = wmma4(a0, bw10_0, yc);
            yc = wmma4(a1, bw10_1, yc);
        }
        #pragma unroll
        for (int r = 0; r < 8; ++r) yc[r] = fmaxf(yc[r] + bb10, 0.0f);

        // C->A re-layout through wave-private LDS slab (same-wave DS is in-order)
        #pragma unroll
        for (int r = 0; r < 8; ++r) myslab[(r + half * 8) * SLABW + n] = yc[r];
        asm volatile("" ::: "memory");

        // ===== scalar branch, stage 2: [16x10] @ [10x10], K padded to 12 =====
        v8f y2 = {};
        #pragma unroll
        for (int kb = 0; kb < 3; ++kb) {
            v2f a;
            const int k0 = kb * 4 + k2;
            a.x = myslab[n * SLABW + k0];       // k>=10 garbage * zero B rows
            a.y = myslab[n * SLABW + k0 + 1];
            y2 = wmma4(a, bw11[kb], y2);
        }
        #pragma unroll
        for (int r = 0; r < 8; ++r) y2[r] = fmaxf(y2[r] + bb11, 0.0f);

        // ===== equivariant stage 1: per component c, [16x3] @ [3x10] =====
        v8f x[3];
        #pragma unroll
        for (int c = 0; c < 3; ++c) {
            v2f a;
            a.x = vvp[n * 9 + k2 * 3 + c];             // u = 0|2, always valid
            a.y = vvp[n * 9 + 3 + c] * hmask;          // u = 1 (half0), u = 3 -> 0
            v8f z = {};
            x[c] = wmma4(a, blin0, z);
        }
        // NormActivation 0: s = tanh(|y| + b) / |y|, via rsq (1/|y| = rsqrt(n2))
        #pragma unroll
        for (int r = 0; r < 8; ++r) {
            float n2 = fmaxf(x[0][r]*x[0][r] + x[1][r]*x[1][r] + x[2][r]*x[2][r], EPS2);
            float rs = __builtin_amdgcn_rsqf(n2);
            float s  = fast_tanh(n2 * rs + nb0) * rs;
            x[0][r] *= s; x[1][r] *= s; x[2][r] *= s;
        }

        // re-layout 3 component matrices to LDS
        #pragma unroll
        for (int c = 0; c < 3; ++c) {
            float* sl = myslab + c * SLABSZ;
            #pragma unroll
            for (int r = 0; r < 8; ++r) sl[(r + half * 8) * SLABW + n] = x[c][r];
        }
        asm volatile("" ::: "memory");

        // ===== equivariant stage 2: per component, [16x10] @ [10x10] =====
        v8f x2[3];
        #pragma unroll
        for (int c = 0; c < 3; ++c) {
            const float* sl = myslab + c * SLABSZ;
            v8f z = {};
            #pragma unroll
            for (int kb = 0; kb < 3; ++kb) {
                v2f a;
                const int k0 = kb * 4 + k2;
                a.x = sl[n * SLABW + k0];
                a.y = sl[n * SLABW + k0 + 1];
                z = wmma4(a, blin1[kb], z);
            }
            x2[c] = z;
        }

        // NormActivation 1 + scalar gate + final projection + pair-sum, fused:
        // out_c += sum_{m,n} x_c[m,n] * y[m,n]*s[m,n] * fw[n]/sqrt(10)
        #pragma unroll
        for (int r = 0; r < 8; ++r) {
            float n2 = fmaxf(x2[0][r]*x2[0][r] + x2[1][r]*x2[1][r] + x2[2][r]*x2[2][r], EPS2);
            float rs = __builtin_amdgcn_rsqf(n2);
            float s  = fast_tanh(n2 * rs + nb1) * rs;
            float g  = y2[r] * s * coef;
            acc0 += g * x2[0][r];
            acc1 += g * x2[1][r];
            acc2 += g * x2[2][r];
        }
    }

    // wave-level reduction (32 lanes)
    #pragma unroll
    for (int off = 16; off >= 1; off >>= 1) {
        acc0 += __shfl_xor(acc0, off, 32);
        acc1 += __shfl_xor(acc1, off, 32);
        acc2 += __shfl_xor(acc2, off, 32);
    }
    if (l == 0) { red[wv][0] = acc0; red[wv][1] = acc1; red[wv][2] = acc2; }
    __syncthreads();
    if (tid < 3) {
        float s = 0.0f;
        #pragma unroll
        for (int w = 0; w < 8; ++w) s += red[w][tid];
        out[(size_t)b * 3 + tid] = s;
    }
}

extern "C" void kernel_launch(void* const* d_in, const int* in_sizes, int n_in,
                              void* d_out, int out_size, void* d_ws, size_t ws_size,
                              hipStream_t stream) {
    const float* invar  = (const float*)d_in[0];
    const float* vecs   = (const float*)d_in[1];
    const float* w1_0   = (const float*)d_in[2];
    const float* b1_0   = (const float*)d_in[3];
    const float* w1_1   = (const float*)d_in[4];
    const float* b1_1   = (const float*)d_in[5];
    const float* lin0_w = (const float*)d_in[6];
    const float* na0_b  = (const float*)d_in[7];
    const float* lin1_w = (const float*)d_in[8];
    const float* na1_b  = (const float*)d_in[9];
    const float* fw     = (const float*)d_in[10];
    float* out = (float*)d_out;

    frc2b_kernel<<<NB, 256, 0, stream>>>(invar, vecs, w1_0, b1_0, w1_1, b1_1,
                                         lin0_w, na0_b, lin1_w, na1_b, fw, out);
}